// TLayer_79010218377405
// MI455X (gfx1250) — compile-verified
//
#include <hip/hip_runtime.h>
#include <hip/hip_bf16.h>
#include <stddef.h>

typedef float v2f __attribute__((ext_vector_type(2)));
typedef float v8f __attribute__((ext_vector_type(8)));

#define TT        2048   // time steps
#define NIN       4096   // input units
#define NLAG      32     // lags (K)
#define NFILT     16     // filters (N of GEMM)
#define WAVES_PB  8
#define T_PER_WV  16
#define NB_BLOCKS (NIN / (16 * WAVES_PB))   // 32 n-blocks
#define TB_BLOCKS (TT / T_PER_WV)           // 128 t-blocks

// out[t, n*16+f] = relu( sum_j x[t-1-2j, n] * w_pn[j, f] + bias[f] )
// GEMM per (t, n-tile): M=n(16) x K=lags(32) x N=f(16) via 8x V_WMMA_F32_16X16X4_F32.
__global__ void __launch_bounds__(256)
tconv16_wmma(const float* __restrict__ x, const float* __restrict__ w,
             const float* __restrict__ bias, float* __restrict__ out)
{
    __shared__ float s_inv[NFILT];
    __shared__ float s_wpn[NLAG * NFILT];

    const int tid = threadIdx.x;

    // weight preprocessing: clip-at-0 then L2-normalize each filter over lags
    if (tid < NFILT) {
        float s = 0.f;
        #pragma unroll
        for (int j = 0; j < NLAG; ++j) {
            float v = fmaxf(w[j * NFILT + tid], 0.f);
            s += v * v;
        }
        s_inv[tid] = rsqrtf(fmaxf(s, 1e-12f));
    }
    __syncthreads();
    for (int i = tid; i < NLAG * NFILT; i += 256)
        s_wpn[i] = fmaxf(w[i], 0.f) * s_inv[i & (NFILT - 1)];
    __syncthreads();

    const int lane = tid & 31;
    const int wave = tid >> 5;
    const int half = lane >> 4;   // 0: lanes 0-15, 1: lanes 16-31
    const int lc   = lane & 15;

    const int tb = blockIdx.x / NB_BLOCKS;
    const int nb = blockIdx.x % NB_BLOCKS;
    const int n0 = (nb * WAVES_PB + wave) * 16;

    // B matrix (w_pn): VGPR v, lane l -> K = 4c + 2*(l>>4) + v, N = l&15
    v2f b[8];
    #pragma unroll
    for (int c = 0; c < 8; ++c) {
        const int k = 4 * c + 2 * half;
        b[c].x = s_wpn[(k + 0) * NFILT + lc];
        b[c].y = s_wpn[(k + 1) * NFILT + lc];
    }
    const float bf = bias[lc];

    // A column base: A[m, k] = x[t-1-2k, n0+m]; both lane halves use m = lc
    const float* __restrict__ xc = x + (n0 + lc);
    const int t_base = tb * T_PER_WV;
    float* orow = out + (size_t)t_base * (NIN * NFILT) + (size_t)n0 * NFILT + lc;

    if (t_base >= 64) {
        // -------- interior fast path: all lag rows >= 1, no predication --------
        // row(c, v) = (t - 1 - 4*half) - 8c - 2v; constant byte offsets from one base
        const float* pb = xc + (size_t)(t_base - 1 - 4 * half) * NIN;
        for (int ti = 0; ti < T_PER_WV; ++ti) {
            float a[16];
            #pragma unroll
            for (int c = 0; c < 8; ++c) {
                a[2 * c + 0] = pb[-(ptrdiff_t)(8 * c + 0) * NIN]; // K = 4c+2h
                a[2 * c + 1] = pb[-(ptrdiff_t)(8 * c + 2) * NIN]; // K = 4c+2h+1
            }
            v8f acc;
            #pragma unroll
            for (int r = 0; r < 8; ++r) acc[r] = bf;   // C = bias broadcast (f = lc)
            #pragma unroll
            for (int c = 0; c < 8; ++c) {
                v2f av; av.x = a[2 * c]; av.y = a[2 * c + 1];
                acc = __builtin_amdgcn_wmma_f32_16x16x4_f32(
                    false, av, false, b[c], (short)0, acc, false, false);
            }
            #pragma unroll
            for (int r = 0; r < 8; ++r)
                __builtin_nontemporal_store(fmaxf(acc[r], 0.f),
                                            orow + (size_t)(r + 8 * half) * NFILT);
            pb   += NIN;
            orow += (size_t)NIN * NFILT;
        }
    } else {
        // -------- boundary path (t < 64): branch-free masking, EXEC stays full --------
        for (int ti = 0; ti < T_PER_WV; ++ti) {
            const int t = t_base + ti;
            v8f acc;
            #pragma unroll
            for (int r = 0; r < 8; ++r) acc[r] = bf;
            #pragma unroll
            for (int c = 0; c < 8; ++c) {
                const int k  = 4 * c + 2 * half;
                const int r0 = t - 1 - 2 * k;
                const int r1 = r0 - 2;
                float v0 = xc[(size_t)(r0 > 0 ? r0 : 0) * NIN]; // clamped, always valid
                float v1 = xc[(size_t)(r1 > 0 ? r1 : 0) * NIN];
                v2f av;
                av.x = (r0 >= 0) ? v0 : 0.f;  // v_cndmask, no EXEC branch
                av.y = (r1 >= 0) ? v1 : 0.f;
                acc = __builtin_amdgcn_wmma_f32_16x16x4_f32(
                    false, av, false, b[c], (short)0, acc, false, false);
            }
            #pragma unroll
            for (int r = 0; r < 8; ++r)
                __builtin_nontemporal_store(fmaxf(acc[r], 0.f),
                                            orow + (size_t)(r + 8 * half) * NFILT);
            orow += (size_t)NIN * NFILT;
        }
    }
}

extern "C" void kernel_launch(void* const* d_in, const int* in_sizes, int n_in,
                              void* d_out, int out_size, void* d_ws, size_t ws_size,
                              hipStream_t stream) {
    (void)in_sizes; (void)n_in; (void)out_size; (void)d_ws; (void)ws_size;
    const float* x    = (const float*)d_in[0];   // [2048, 4096]
    const float* w    = (const float*)d_in[1];   // [32, 16]
    const float* bias = (const float*)d_in[2];   // [16]
    float* out = (float*)d_out;                  // [2048, 65536]

    dim3 grid(TB_BLOCKS * NB_BLOCKS);            // 4096 blocks
    dim3 block(256);                             // 8 waves/block
    tconv16_wmma<<<grid, block, 0, stream>>>(x, w, bias, out);
}